// FeatEmb_69406671503913
// MI455X (gfx1250) — compile-verified
//
#include <hip/hip_runtime.h>
#include <cstddef>

// ---------------------------------------------------------------------------
// CDNA5 / gfx1250 implementation of the DGCNN feature-embedding block.
// All matmuls (edge-conv 1x1 convs, FC chain, KNN similarity) are driven by
// V_WMMA_F32_16X16X4_F32 to stay bit-faithful to the fp32 reference while
// still using the matrix pipes. wave32, one wave per workgroup.
// GEMM: 32x64 output tile per wave; per k-step the 4 B fragments are loaded
// as a clustered burst before the 8 WMMAs so loads stay in flight.
// ---------------------------------------------------------------------------

typedef __attribute__((ext_vector_type(2))) float v2f;
typedef __attribute__((ext_vector_type(8))) float v8f;

#define NEG_INF (-__builtin_inff())
#define BN_EPS 1e-5f

__device__ __forceinline__ v8f wmma4(v2f a, v2f b, v8f c) {
    // D = A(16x4) x B(4x16) + C(16x16), fp32 exact.
    return __builtin_amdgcn_wmma_f32_16x16x4_f32(
        /*neg_a=*/false, a, /*neg_b=*/false, b,
        /*c_mod=*/(short)0, c, /*reuse_a=*/false, /*reuse_b=*/false);
}

// Fragment layout (per CDNA5 ISA):
//  A 16x4 : lanes 0-15 hold K=0,1 ; lanes 16-31 hold K=2,3 (row = lane&15)
//  B 4x16 : lanes 0-15 hold K=0,1 ; lanes 16-31 hold K=2,3 (col = lane&15)
//  C 16x16: c[v] -> row = v + 8*(lane>=16), col = lane&15

// ---------------------------------------------------------------------------
// Generic GEMM: out[M,O] = act( X[M,K] * W[O,K]^T ), 32x64 tile per wave.
// Requires M % 32 == 0, O % 64 == 0, K % 4 == 0 (true for every layer here).
// act: 0=none, 1=eval-BN + LeakyReLU(0.2), 2=+bias
// ---------------------------------------------------------------------------
__global__ __launch_bounds__(32) void gemm_wt_kernel(
    const float* __restrict__ X, const float* __restrict__ W,
    float* __restrict__ out, int M, int Kdim, int O, int act,
    const float* __restrict__ g, const float* __restrict__ bb,
    const float* __restrict__ mm, const float* __restrict__ vv,
    const float* __restrict__ bias)
{
    const int o0 = blockIdx.x * 64;
    const int m0 = blockIdx.y * 32;
    const int lane = threadIdx.x;
    const int half = lane >> 4;
    const int l16  = lane & 15;

    const float* xrow0 = X + (size_t)(m0 + l16) * Kdim;
    const float* xrow1 = xrow0 + (size_t)16 * Kdim;
    const float* wr0 = W + (size_t)(o0 + l16) * Kdim;
    const float* wr1 = wr0 + (size_t)16 * Kdim;
    const float* wr2 = wr0 + (size_t)32 * Kdim;
    const float* wr3 = wr0 + (size_t)48 * Kdim;

    v8f acc[2][4] = {};
    for (int k0 = 0; k0 < Kdim; k0 += 4) {
        const int ka = k0 + half * 2;
        // burst all 6 fragment loads, then fire the 8 WMMAs
        v2f a0; a0.x = xrow0[ka]; a0.y = xrow0[ka + 1];
        v2f a1; a1.x = xrow1[ka]; a1.y = xrow1[ka + 1];
        v2f bf[4];
        bf[0].x = wr0[ka]; bf[0].y = wr0[ka + 1];
        bf[1].x = wr1[ka]; bf[1].y = wr1[ka + 1];
        bf[2].x = wr2[ka]; bf[2].y = wr2[ka + 1];
        bf[3].x = wr3[ka]; bf[3].y = wr3[ka + 1];
#pragma unroll
        for (int j = 0; j < 4; ++j) {
            acc[0][j] = wmma4(a0, bf[j], acc[0][j]);
            acc[1][j] = wmma4(a1, bf[j], acc[1][j]);
        }
    }

#pragma unroll
    for (int j = 0; j < 4; ++j) {
        const int col = o0 + j * 16 + l16;
        float scale = 1.f, shift = 0.f;
        if (act == 1) {
            const float iv = rsqrtf(vv[col] + BN_EPS);
            scale = g[col] * iv;
            shift = bb[col] - mm[col] * scale;
        } else if (act == 2) {
            shift = bias[col];
        }
#pragma unroll
        for (int r = 0; r < 2; ++r) {
            float* orow = out + (size_t)(m0 + r * 16 + 8 * half) * O + col;
#pragma unroll
            for (int v = 0; v < 8; ++v) {
                float val = acc[r][j][v];
                if (act == 1) { val = val * scale + shift; val = val > 0.f ? val : 0.2f * val; }
                else if (act == 2) { val += shift; }
                orow[(size_t)v * O] = val;
            }
        }
    }
}

// ---------------------------------------------------------------------------
// Edge conv: one wave per point. 20 neighbor rows -> two 16-row tiles.
// e = [neigh-center, center] (2*DIN) ; h1 = lrelu(bn(e W1^T)) (2*DOUT);
// h2 = h1 W2^T + b2 (DOUT) ; f1[n] = max over valid k. All in-wave, no atomics.
// ---------------------------------------------------------------------------
template <int DIN, int DOUT>
__global__ __launch_bounds__(32) void edgeconv_kernel(
    const float* __restrict__ feat,   // [N, DIN]
    const int*   __restrict__ knn,    // [N, 20]
    const float* __restrict__ W1,     // [2*DOUT, 2*DIN]
    const float* __restrict__ g1, const float* __restrict__ b1,
    const float* __restrict__ m1, const float* __restrict__ v1,
    const float* __restrict__ W2,     // [DOUT, 2*DOUT]
    const float* __restrict__ b2,
    float* __restrict__ f1)           // [N, DOUT]
{
    constexpr int C = 2 * DIN;    // e row length
    constexpr int H = 2 * DOUT;   // hidden width
    constexpr int OT = DOUT / 16; // output col tiles

    __shared__ float Etile[16][C + 1];  // +1: bank-conflict-free column reads
    __shared__ float Stg[16][17];       // D-layout -> A-layout re-fragment

    const int n    = blockIdx.x;
    const int lane = threadIdx.x;
    const int half = lane >> 4;
    const int l16  = lane & 15;
    const float* center = feat + (size_t)n * DIN;

    float fmaxcol[OT];
#pragma unroll
    for (int o = 0; o < OT; ++o) fmaxcol[o] = NEG_INF;

    for (int t = 0; t < 2; ++t) {
        // ---- build E tile (rows k = 16t .. 16t+15, zero-pad k>=20) ----
        for (int j = 0; j < 16; ++j) {
            const int k = t * 16 + j;
            if (k < 20) {
                const int nb = knn[n * 20 + k];
                const float* neigh = feat + (size_t)nb * DIN;
                for (int c = lane; c < C; c += 32)
                    Etile[j][c] = (c < DIN) ? (neigh[c] - center[c]) : center[c - DIN];
            } else {
                for (int c = lane; c < C; c += 32) Etile[j][c] = 0.f;
            }
        }
        __syncthreads();

        v8f acc2[OT] = {};

        // ---- loop over hidden column chunks; fuse 2nd GEMM ----
        for (int c0 = 0; c0 < H; c0 += 16) {
            v8f h = {};
            const float* w1row = W1 + (size_t)(c0 + l16) * C;
            for (int k0 = 0; k0 < C; k0 += 4) {
                const int ka = k0 + half * 2;
                v2f a; a.x = Etile[l16][ka]; a.y = Etile[l16][ka + 1];
                v2f b; b.x = w1row[ka];      b.y = w1row[ka + 1];
                h = wmma4(a, b, h);
            }
            // BN (eval) + LeakyReLU, stage to LDS in C-layout
            const int col = c0 + l16;
            const float iv = rsqrtf(v1[col] + BN_EPS);
            const float sc = g1[col] * iv;
            const float sh = b1[col] - m1[col] * sc;
#pragma unroll
            for (int v = 0; v < 8; ++v) {
                float val = h[v] * sc + sh;
                val = val > 0.f ? val : 0.2f * val;
                Stg[v + 8 * half][l16] = val;
            }
            __syncthreads();
            // 2nd GEMM partial: K-slice = this 16-wide h1 chunk.
            // Burst the OT weight fragments per k-step, then fire WMMAs.
#pragma unroll
            for (int kk = 0; kk < 4; ++kk) {
                const int ka = kk * 4 + half * 2;
                v2f a2; a2.x = Stg[l16][ka]; a2.y = Stg[l16][ka + 1];
                v2f bfr[OT];
#pragma unroll
                for (int o = 0; o < OT; ++o) {
                    const float* w2row = W2 + (size_t)(o * 16 + l16) * H;
                    bfr[o].x = w2row[c0 + ka]; bfr[o].y = w2row[c0 + ka + 1];
                }
#pragma unroll
                for (int o = 0; o < OT; ++o)
                    acc2[o] = wmma4(a2, bfr[o], acc2[o]);
            }
            __syncthreads();
        }

        // ---- bias + masked max over valid k rows ----
#pragma unroll
        for (int o = 0; o < OT; ++o) {
            const float bsv = b2[o * 16 + l16];
#pragma unroll
            for (int v = 0; v < 8; ++v) {
                const int k = t * 16 + v + 8 * half;
                if (k < 20) fmaxcol[o] = fmaxf(fmaxcol[o], acc2[o][v] + bsv);
            }
        }
        __syncthreads();
    }

    // combine the two lane halves (cols are lane&15 in both) and store f1[n]
#pragma unroll
    for (int o = 0; o < OT; ++o) {
        const float other = __shfl_xor(fmaxcol[o], 16, 32);
        const float r = fmaxf(fmaxcol[o], other);
        if (half == 0) f1[(size_t)n * DOUT + o * 16 + l16] = r;
    }
}

// ---------------------------------------------------------------------------
// Global max/avg pool over N rows for one column per block.
// ---------------------------------------------------------------------------
__global__ void pool_kernel(const float* __restrict__ f1, int N, int D,
                            float* __restrict__ pool)  // [0..D)=max, [D..2D)=avg
{
    const int c = blockIdx.x;
    __shared__ float smax[256], ssum[256];
    float mx = NEG_INF, sm = 0.f;
    for (int r = threadIdx.x; r < N; r += 256) {
        const float v = f1[(size_t)r * D + c];
        mx = fmaxf(mx, v); sm += v;
    }
    smax[threadIdx.x] = mx; ssum[threadIdx.x] = sm;
    __syncthreads();
    for (int s = 128; s > 0; s >>= 1) {
        if (threadIdx.x < s) {
            smax[threadIdx.x] = fmaxf(smax[threadIdx.x], smax[threadIdx.x + s]);
            ssum[threadIdx.x] += ssum[threadIdx.x + s];
        }
        __syncthreads();
    }
    if (threadIdx.x == 0) { pool[c] = smax[0]; pool[D + c] = ssum[0] / (float)N; }
}

// x = concat[featlike(din), f1(dout), fmax bcast, favg bcast]  -> [N, din+3*dout]
__global__ void concat_kernel(const float* __restrict__ featlike, int din,
                              const float* __restrict__ f1, int dout,
                              const float* __restrict__ pool,
                              float* __restrict__ x, int N)
{
    const int cin = din + 3 * dout;
    const int i = blockIdx.x * blockDim.x + threadIdx.x;
    if (i >= N * cin) return;
    const int nrow = i / cin, c = i % cin;
    float v;
    if (c < din)               v = featlike[(size_t)nrow * din + c];
    else if (c < din + dout)   v = f1[(size_t)nrow * dout + (c - din)];
    else if (c < din + 2*dout) v = pool[c - din - dout];
    else                       v = pool[dout + (c - din - 2 * dout)];
    x[i] = v;
}

__global__ void sqnorm_kernel(const float* __restrict__ F, float* __restrict__ xx,
                              int N, int D)
{
    const int n = blockIdx.x * blockDim.x + threadIdx.x;
    if (n >= N) return;
    float s = 0.f;
    for (int c = 0; c < D; ++c) { const float v = F[(size_t)n * D + c]; s += v * v; }
    xx[n] = s;
}

// ---------------------------------------------------------------------------
// KNN: wave per 16 query rows; WMMA dot tiles; per-lane unrolled sorted top-20
// (score = 2*dot - ||m||^2 ; the row-constant -||n||^2 never changes order).
// ---------------------------------------------------------------------------
__global__ __launch_bounds__(32) void knn_kernel(
    const float* __restrict__ F,   // [N, 64]
    const float* __restrict__ xx,  // [N]
    int* __restrict__ knnout,      // [N, 20]
    int N)
{
    __shared__ float Sc[16][17];
    const int lane = threadIdx.x;
    const int half = lane >> 4;
    const int l16  = lane & 15;
    const int n0   = blockIdx.x * 16;

    // keep all 16 A k-step fragments resident (32 VGPRs)
    v2f afr[16];
    const float* arow = F + (size_t)(n0 + l16) * 64;
#pragma unroll
    for (int s = 0; s < 16; ++s) {
        const int ka = s * 4 + half * 2;
        afr[s].x = arow[ka]; afr[s].y = arow[ka + 1];
    }

    float val[20]; int idxl[20];
#pragma unroll
    for (int j = 0; j < 20; ++j) { val[j] = NEG_INF; idxl[j] = 0; }
    const int myrow = n0 + l16;

    for (int m0 = 0; m0 < N; m0 += 16) {
        v8f c = {};
        const float* brow = F + (size_t)(m0 + l16) * 64;
#pragma unroll
        for (int s = 0; s < 16; ++s) {
            const int ka = s * 4 + half * 2;
            v2f b; b.x = brow[ka]; b.y = brow[ka + 1];
            c = wmma4(afr[s], b, c);
        }
        const float xm = xx[m0 + l16];
#pragma unroll
        for (int v = 0; v < 8; ++v)
            Sc[v + 8 * half][l16] = 2.f * c[v] - xm;
        __syncthreads();

        if (lane < 16) {
#pragma unroll
            for (int j = 0; j < 16; ++j) {
                const int m = m0 + j;
                const float s = Sc[lane][j];
                if (m != myrow && s > val[19]) {     // strict > : low-index tie-break
                    val[19] = s; idxl[19] = m;
#pragma unroll
                    for (int q = 19; q > 0; --q) {   // bubble new min-slot up
                        if (val[q] > val[q - 1]) {
                            const float tv = val[q]; val[q] = val[q - 1]; val[q - 1] = tv;
                            const int   ti = idxl[q]; idxl[q] = idxl[q - 1]; idxl[q - 1] = ti;
                        }
                    }
                }
            }
        }
        __syncthreads();
    }

    if (lane < 16) {
#pragma unroll
        for (int j = 0; j < 20; ++j) knnout[(size_t)myrow * 20 + j] = idxl[j];
    }
}

__global__ void assemble_kernel(const float* __restrict__ out0,
                                const float* __restrict__ out1,
                                float* __restrict__ out, int N)
{
    const int i = blockIdx.x * blockDim.x + threadIdx.x;
    if (i >= N * 192) return;
    const int nrow = i / 192, c = i % 192;
    out[i] = (c < 64) ? out0[(size_t)nrow * 64 + c]
                      : out1[(size_t)nrow * 128 + (c - 64)];
}

// ---------------------------------------------------------------------------
extern "C" void kernel_launch(void* const* d_in, const int* in_sizes, int n_in,
                              void* d_out, int out_size, void* d_ws, size_t ws_size,
                              hipStream_t stream)
{
    const int N = in_sizes[0] / 64;   // 8192
    const float* feat = (const float*)d_in[0];
    const int*   idx0 = (const int*)d_in[1];

    // p0 (din=64, dout=64): ec_w1, ec_bn1{g,b,m,v}, ec_w2, ec_b2,
    //                       fc_w1, fc_bn1{..}, fc_w2, fc_bn2{..}, fc_w3, fc_bn3{..}, fc_w4, fc_b4
    const float** P = (const float**)d_in;
    const float *p0_ecw1 = P[2],  *p0_g1 = P[3],  *p0_b1 = P[4],  *p0_m1 = P[5],  *p0_v1 = P[6];
    const float *p0_ecw2 = P[7],  *p0_ecb2 = P[8];
    const float *p0_fw1 = P[9],   *p0_f1g = P[10], *p0_f1b = P[11], *p0_f1m = P[12], *p0_f1v = P[13];
    const float *p0_fw2 = P[14],  *p0_f2g = P[15], *p0_f2b = P[16], *p0_f2m = P[17], *p0_f2v = P[18];
    const float *p0_fw3 = P[19],  *p0_f3g = P[20], *p0_f3b = P[21], *p0_f3m = P[22], *p0_f3v = P[23];
    const float *p0_fw4 = P[24],  *p0_fb4 = P[25];
    const float *p1_ecw1 = P[26], *p1_g1 = P[27], *p1_b1 = P[28], *p1_m1 = P[29], *p1_v1 = P[30];
    const float *p1_ecw2 = P[31], *p1_ecb2 = P[32];
    const float *p1_fw1 = P[33],  *p1_f1g = P[34], *p1_f1b = P[35], *p1_f1m = P[36], *p1_f1v = P[37];
    const float *p1_fw2 = P[38],  *p1_f2g = P[39], *p1_f2b = P[40], *p1_f2m = P[41], *p1_f2v = P[42];
    const float *p1_fw3 = P[43],  *p1_f3g = P[44], *p1_f3b = P[45], *p1_f3m = P[46], *p1_f3v = P[47];
    const float *p1_fw4 = P[48],  *p1_fb4 = P[49];

    // workspace carve-up (reused across phases)
    float* ws = (float*)d_ws;
    size_t o = 0;
    float* xbuf  = ws + o; o += (size_t)N * 448;
    float* ya    = ws + o; o += (size_t)N * 384;
    float* yb    = ws + o; o += (size_t)N * 256;
    float* f1buf = ws + o; o += (size_t)N * 128;   // f1_0 then f1_1
    float* out0  = ws + o; o += (size_t)N * 64;
    float* out1  = ws + o; o += (size_t)N * 128;
    float* pool  = ws + o; o += 1024;
    float* xxb   = ws + o; o += (size_t)N;
    int*   idx1  = (int*)(ws + o);

    const dim3 wave(32);
    const int MT = N / 32;                  // GEMM row tiles (32 rows/wave)
    const int NT = (N * 448 + 255) / 256;   // generous 1D grid for elementwise

    // ---- block 0 ----
    edgeconv_kernel<64, 64><<<N, wave, 0, stream>>>(
        feat, idx0, p0_ecw1, p0_g1, p0_b1, p0_m1, p0_v1, p0_ecw2, p0_ecb2, f1buf);
    pool_kernel<<<64, 256, 0, stream>>>(f1buf, N, 64, pool);
    concat_kernel<<<(N * 256 + 255) / 256, 256, 0, stream>>>(feat, 64, f1buf, 64, pool, xbuf, N);
    gemm_wt_kernel<<<dim3(192 / 64, MT), wave, 0, stream>>>(
        xbuf, p0_fw1, ya, N, 256, 192, 1, p0_f1g, p0_f1b, p0_f1m, p0_f1v, nullptr);
    gemm_wt_kernel<<<dim3(128 / 64, MT), wave, 0, stream>>>(
        ya, p0_fw2, yb, N, 192, 128, 1, p0_f2g, p0_f2b, p0_f2m, p0_f2v, nullptr);
    gemm_wt_kernel<<<dim3(64 / 64, MT), wave, 0, stream>>>(
        yb, p0_fw3, ya, N, 128, 64, 1, p0_f3g, p0_f3b, p0_f3m, p0_f3v, nullptr);
    gemm_wt_kernel<<<dim3(64 / 64, MT), wave, 0, stream>>>(
        ya, p0_fw4, out0, N, 64, 64, 2, nullptr, nullptr, nullptr, nullptr, p0_fb4);

    // ---- recompute KNN on out0 ----
    sqnorm_kernel<<<(N + 255) / 256, 256, 0, stream>>>(out0, xxb, N, 64);
    knn_kernel<<<N / 16, wave, 0, stream>>>(out0, xxb, idx1, N);

    // ---- block 1 ----
    edgeconv_kernel<64, 128><<<N, wave, 0, stream>>>(
        out0, idx1, p1_ecw1, p1_g1, p1_b1, p1_m1, p1_v1, p1_ecw2, p1_ecb2, f1buf);
    pool_kernel<<<128, 256, 0, stream>>>(f1buf, N, 128, pool);
    concat_kernel<<<NT, 256, 0, stream>>>(out0, 64, f1buf, 128, pool, xbuf, N);
    gemm_wt_kernel<<<dim3(384 / 64, MT), wave, 0, stream>>>(
        xbuf, p1_fw1, ya, N, 448, 384, 1, p1_f1g, p1_f1b, p1_f1m, p1_f1v, nullptr);
    gemm_wt_kernel<<<dim3(256 / 64, MT), wave, 0, stream>>>(
        ya, p1_fw2, yb, N, 384, 256, 1, p1_f2g, p1_f2b, p1_f2m, p1_f2v, nullptr);
    gemm_wt_kernel<<<dim3(128 / 64, MT), wave, 0, stream>>>(
        yb, p1_fw3, ya, N, 256, 128, 1, p1_f3g, p1_f3b, p1_f3m, p1_f3v, nullptr);
    gemm_wt_kernel<<<dim3(128 / 64, MT), wave, 0, stream>>>(
        ya, p1_fw4, out1, N, 128, 128, 2, nullptr, nullptr, nullptr, nullptr, p1_fb4);

    // ---- output [N, 64+128] ----
    assemble_kernel<<<(N * 192 + 255) / 256, 256, 0, stream>>>(out0, out1, (float*)d_out, N);
}